// GraphNetLowCapacity_83614423318871
// MI455X (gfx1250) — compile-verified
//
#include <hip/hip_runtime.h>
#include <hip/hip_bf16.h>
#include <math.h>

typedef __attribute__((ext_vector_type(2))) float v2f;
typedef __attribute__((ext_vector_type(8))) float v8f;

#define EPS_BN 1e-5f
#define NUM_G 512

// ---------------- degree / normalization ----------------
__global__ void k_deg_init(float* deg, int n) {
  int i = blockIdx.x * blockDim.x + threadIdx.x;
  if (i < n) deg[i] = 1.0f;                 // self-loop
}
__global__ void k_deg_count(const int* __restrict__ dst, float* deg, int e) {
  int i = blockIdx.x * blockDim.x + threadIdx.x;
  if (i < e) atomicAdd(&deg[dst[i]], 1.0f);
}
__global__ void k_deg_rsqrt(float* deg, int n) {
  int i = blockIdx.x * blockDim.x + threadIdx.x;
  if (i < n) deg[i] = rsqrtf(fmaxf(deg[i], 1.0f));
}

// ---------------- zero-padding for layer 1 (K=6 -> K=8) ----------------
__global__ void k_pad_x(const float* __restrict__ x, float* __restrict__ xp, int n) {
  int i = blockIdx.x * blockDim.x + threadIdx.x;   // over n*8
  if (i >= n * 8) return;
  int node = i >> 3, k = i & 7;
  xp[i] = (k < 6) ? x[node * 6 + k] : 0.0f;
}
__global__ void k_pad_w1(const float* __restrict__ w, float* __restrict__ wp) {
  int i = blockIdx.x * blockDim.x + threadIdx.x;   // over 8*32
  if (i >= 8 * 32) return;
  int k = i >> 5;
  wp[i] = (k < 6) ? w[i] : 0.0f;                   // row-major [K,32], rows 6,7 = 0
}

// ---------------- fp32 WMMA GEMM: H[M,Nc] = A[M,K] @ B[K,Nc] ----------------
// Compile-time K (multiple of 4) and Nc (multiple of 16). One wave per 16x16
// tile, fully unrolled chain of V_WMMA_F32_16X16X4_F32.
template <int K, int Nc>
__global__ void k_gemm_wmma(const float* __restrict__ A, const float* __restrict__ B,
                            float* __restrict__ H, int M) {
  static_assert((K & 3) == 0 && (Nc & 15) == 0, "shape");
  const int wave = blockIdx.x * (blockDim.x >> 5) + (threadIdx.x >> 5);
  const int lane = threadIdx.x & 31;
  constexpr int tilesN = Nc >> 4;
  const int tiles = (M >> 4) * tilesN;
  if (wave >= tiles) return;                       // wave-uniform exit
  const int tm = wave / tilesN;
  const int tn = wave % tilesN;                    // constexpr pow2 -> shifts
  const int half = lane >> 4;                      // lanes 0-15 / 16-31
  const int l16  = lane & 15;
  const int row  = (tm << 4) + l16;
  const int col  = (tn << 4) + l16;
  const float* __restrict__ Arow = A + (size_t)row * K;
  v8f c = {};
#pragma unroll
  for (int k0 = 0; k0 < K; k0 += 4) {
    const int ka = k0 + (half << 1);               // this lane-half: K = ka, ka+1
    v2f a = *reinterpret_cast<const v2f*>(Arow + ka);   // 8B aligned (ka even)
    v2f b;
    b.x = B[(size_t)(ka)     * Nc + col];
    b.y = B[(size_t)(ka + 1) * Nc + col];
    c = __builtin_amdgcn_wmma_f32_16x16x4_f32(false, a, false, b, (short)0, c,
                                              false, false);
  }
  const int rbase = (tm << 4) + (half << 3);
#pragma unroll
  for (int r = 0; r < 8; ++r)
    H[(size_t)(rbase + r) * Nc + col] = c[r];
}

// ---------------- aggregation ----------------
// agg[n,:] = dinv[n]^2 * h[n,:]   (self-loop term, also serves as init)
__global__ void k_agg_init(const float* __restrict__ h, const float* __restrict__ dinv,
                           float* __restrict__ agg, int n, int cshift) {
  int i = blockIdx.x * blockDim.x + threadIdx.x;
  int total = n << cshift;
  if (i >= total) return;
  int node = i >> cshift;
  float w = dinv[node];
  agg[i] = w * w * h[i];
}

// edge scatter: agg[d,:] += dinv[s]*dinv[d] * h[s,:]; 4 channels per thread
__global__ void k_scatter(const int* __restrict__ src, const int* __restrict__ dst,
                          const float* __restrict__ dinv, const float* __restrict__ h,
                          float* __restrict__ agg, int e, int C, int chunkShift) {
  int i = blockIdx.x * blockDim.x + threadIdx.x;
  int total = e << chunkShift;                     // chunks of 4 channels
  if (i >= total) return;
  int eidx = i >> chunkShift;
  int cc = (i & ((1 << chunkShift) - 1)) << 2;
  int s = src[eidx], d = dst[eidx];
  float w = dinv[s] * dinv[d];
  const float4 hv = *reinterpret_cast<const float4*>(h + (size_t)s * C + cc);
  float* p = agg + (size_t)d * C + cc;
  atomicAdd(p + 0, w * hv.x);
  atomicAdd(p + 1, w * hv.y);
  atomicAdd(p + 2, w * hv.z);
  atomicAdd(p + 3, w * hv.w);
}

// ---------------- BatchNorm (training stats) ----------------
// one block per channel (deterministic tree reduction, no atomics)
__global__ void k_bnstats(const float* __restrict__ x, float* __restrict__ stats,
                          int n, int C) {
  int c = blockIdx.x;
  float s = 0.0f, s2 = 0.0f;
  for (int i = threadIdx.x; i < n; i += blockDim.x) {
    float v = x[(size_t)i * C + c];
    s += v; s2 += v * v;
  }
  __shared__ float sh[256], sh2[256];
  sh[threadIdx.x] = s; sh2[threadIdx.x] = s2;
  __syncthreads();
  for (int w = blockDim.x >> 1; w > 0; w >>= 1) {
    if ((int)threadIdx.x < w) {
      sh[threadIdx.x]  += sh[threadIdx.x + w];
      sh2[threadIdx.x] += sh2[threadIdx.x + w];
    }
    __syncthreads();
  }
  if (threadIdx.x == 0) { stats[c] = sh[0]; stats[C + c] = sh2[0]; }
}

// in-place y = relu(g*(x-mean)*rsqrt(var+eps)+be); C is a power of two
__global__ void k_bnrelu(float* __restrict__ x, const float* __restrict__ stats,
                         const float* __restrict__ g, const float* __restrict__ be,
                         int n, int C) {
  int i = blockIdx.x * blockDim.x + threadIdx.x;
  int total = n * C;
  if (i >= total) return;
  int c = i & (C - 1);
  float inv_n = 1.0f / (float)n;
  float m = stats[c] * inv_n;
  float var = stats[C + c] * inv_n - m * m;
  float sc = g[c] * rsqrtf(var + EPS_BN);
  float v = (x[i] - m) * sc + be[c];
  x[i] = fmaxf(v, 0.0f);
}

// ---------------- per-graph max pooling ----------------
__global__ void k_pool_init(unsigned* pool, int total) {
  int i = blockIdx.x * blockDim.x + threadIdx.x;
  if (i < total) pool[i] = 0u;                     // 0.0f bits; inputs post-ReLU >= 0
}
__global__ void k_pool_max(const float* __restrict__ h, const int* __restrict__ batch,
                           unsigned* __restrict__ pool, int n, int cshift) {
  int i = blockIdx.x * blockDim.x + threadIdx.x;
  int total = n << cshift;
  if (i >= total) return;
  int node = i >> cshift;
  int c = i & ((1 << cshift) - 1);
  int g = batch[node];
  // bit pattern of non-negative floats is order-preserving as unsigned
  atomicMax(&pool[((size_t)g << cshift) + c], __float_as_uint(h[i]));
}

// ---------------- final bias + L2 normalize (one block per graph) ----------------
__global__ void k_finalize(const float* __restrict__ r, const float* __restrict__ bf2,
                           float* __restrict__ out) {
  int g = blockIdx.x;
  int c = threadIdx.x;                             // 0..63
  float v = r[(size_t)g * 64 + c] + bf2[c];
  __shared__ float sh[64];
  sh[c] = v * v;
  __syncthreads();
  for (int w = 32; w > 0; w >>= 1) {
    if (c < w) sh[c] += sh[c + w];
    __syncthreads();
  }
  float nrm = sqrtf(sh[0]);
  out[(size_t)g * 64 + c] = v / fmaxf(nrm, 1e-12f);
}

// ---------------- host side ----------------
template <int K, int Nc>
static inline void launch_gemm(const float* A, const float* B, float* H,
                               int M, hipStream_t s) {
  int tiles = (M >> 4) * (Nc >> 4);
  int blocks = (tiles + 7) / 8;                    // 256 threads = 8 waves/block
  k_gemm_wmma<K, Nc><<<blocks, 256, 0, s>>>(A, B, H, M);
}

extern "C" void kernel_launch(void* const* d_in, const int* in_sizes, int n_in,
                              void* d_out, int out_size, void* d_ws, size_t ws_size,
                              hipStream_t stream) {
  const float* x     = (const float*)d_in[0];
  const int*   esrc  = (const int*)  d_in[1];
  const int*   edst  = (const int*)  d_in[2];
  const int*   batch = (const int*)  d_in[3];
  const float* W1  = (const float*)d_in[4];
  const float* g1  = (const float*)d_in[6];
  const float* be1 = (const float*)d_in[7];
  const float* W2  = (const float*)d_in[8];
  const float* g2  = (const float*)d_in[10];
  const float* be2 = (const float*)d_in[11];
  const float* W3  = (const float*)d_in[12];
  const float* g3  = (const float*)d_in[14];
  const float* be3 = (const float*)d_in[15];
  const float* Wf1 = (const float*)d_in[16];
  const float* g4  = (const float*)d_in[18];
  const float* be4 = (const float*)d_in[19];
  const float* Wf2 = (const float*)d_in[20];
  const float* bf2 = (const float*)d_in[21];
  // b1/b2/b3/bf1 are shift-invariant under the following BatchNorm -> skipped.

  const int N = in_sizes[0] / 6;                   // 100000 (multiple of 16)
  const int E = in_sizes[1];

  // workspace layout (floats), keep 16B alignment (N rounded to 4)
  size_t Np = ((size_t)N + 3) & ~(size_t)3;
  float* ws     = (float*)d_ws;
  float* dinv   = ws;                              // Np
  float* hbuf   = dinv + Np;                       // Np*128
  float* agg    = hbuf + Np * 128;                 // Np*128
  float* xpad   = agg + Np * 128;                  // Np*8
  float* w1pad  = xpad + Np * 8;                   // 256
  float* stats  = w1pad + 256;                     // 256
  float* pool   = stats + 256;                     // 512*128 (uint bits of floats)
  float* q      = pool + (size_t)NUM_G * 128;      // 512*128
  float* r      = q + (size_t)NUM_G * 128;         // 512*64
  float* stats2 = r + (size_t)NUM_G * 64;          // 256

  const int TB = 256;
  // degrees -> dinv
  k_deg_init<<<(N + TB - 1) / TB, TB, 0, stream>>>(dinv, N);
  k_deg_count<<<(E + TB - 1) / TB, TB, 0, stream>>>(edst, dinv, E);
  k_deg_rsqrt<<<(N + TB - 1) / TB, TB, 0, stream>>>(dinv, N);

  // pad layer-1 operands to K=8
  k_pad_x<<<((N * 8) + TB - 1) / TB, TB, 0, stream>>>(x, xpad, N);
  k_pad_w1<<<1, TB, 0, stream>>>(W1, w1pad);

  // ---- layer 1: 6(->8) -> 32 ----
  launch_gemm<8, 32>(xpad, w1pad, hbuf, N, stream);
  k_agg_init<<<((N << 5) + TB - 1) / TB, TB, 0, stream>>>(hbuf, dinv, agg, N, 5);
  k_scatter<<<((E << 3) + TB - 1) / TB, TB, 0, stream>>>(esrc, edst, dinv, hbuf, agg, E, 32, 3);
  k_bnstats<<<32, TB, 0, stream>>>(agg, stats, N, 32);
  k_bnrelu<<<((N << 5) + TB - 1) / TB, TB, 0, stream>>>(agg, stats, g1, be1, N, 32);

  // ---- layer 2: 32 -> 64 ----
  launch_gemm<32, 64>(agg, W2, hbuf, N, stream);
  k_agg_init<<<((N << 6) + TB - 1) / TB, TB, 0, stream>>>(hbuf, dinv, agg, N, 6);
  k_scatter<<<((E << 4) + TB - 1) / TB, TB, 0, stream>>>(esrc, edst, dinv, hbuf, agg, E, 64, 4);
  k_bnstats<<<64, TB, 0, stream>>>(agg, stats, N, 64);
  k_bnrelu<<<((N << 6) + TB - 1) / TB, TB, 0, stream>>>(agg, stats, g2, be2, N, 64);

  // ---- layer 3: 64 -> 128 ----
  launch_gemm<64, 128>(agg, W3, hbuf, N, stream);
  k_agg_init<<<((N << 7) + TB - 1) / TB, TB, 0, stream>>>(hbuf, dinv, agg, N, 7);
  k_scatter<<<((E << 5) + TB - 1) / TB, TB, 0, stream>>>(esrc, edst, dinv, hbuf, agg, E, 128, 5);
  k_bnstats<<<128, TB, 0, stream>>>(agg, stats, N, 128);
  k_bnrelu<<<((N << 7) + TB - 1) / TB, TB, 0, stream>>>(agg, stats, g3, be3, N, 128);

  // ---- per-graph max pool (post-ReLU values are >= 0) ----
  k_pool_init<<<((NUM_G * 128) + TB - 1) / TB, TB, 0, stream>>>((unsigned*)pool, NUM_G * 128);
  k_pool_max<<<((N << 7) + TB - 1) / TB, TB, 0, stream>>>(agg, batch, (unsigned*)pool, N, 7);

  // ---- FC1 128->128 + BN + ReLU (bf1 cancels in BN) ----
  launch_gemm<128, 128>(pool, Wf1, q, NUM_G, stream);
  k_bnstats<<<128, TB, 0, stream>>>(q, stats2, NUM_G, 128);
  k_bnrelu<<<((NUM_G * 128) + TB - 1) / TB, TB, 0, stream>>>(q, stats2, g4, be4, NUM_G, 128);

  // ---- FC2 128->64 + bias + row L2-normalize ----
  launch_gemm<128, 64>(q, Wf2, r, NUM_G, stream);
  k_finalize<<<NUM_G, 64, 0, stream>>>(r, bf2, (float*)d_out);
}